// RCOVGINModel_77541339562356
// MI455X (gfx1250) — compile-verified
//
#include <hip/hip_runtime.h>
#include <hip/hip_bf16.h>

#define N_NODES   50000
#define N_EDGES   1600000
#define HID       128
#define N_LAYERS  3
#define N_GRAPHS  128
#define N_CLASSES 10
#define N_PAD     50048          // next multiple of 128
#define LDSW      130            // 128 + 2 pad floats: breaks 64-bank stride conflicts

typedef __attribute__((ext_vector_type(2))) float v2f;
typedef __attribute__((ext_vector_type(8))) float v8f;

// Non-returning fp32 atomic add: relaxed, device (agent) scope.
// Result is dead -> backend emits global_atomic_add_f32 without return
// (STOREcnt-tracked), avoiding any CAS fallback.
__device__ __forceinline__ void atomic_add_f32(float* p, float v) {
  (void)__hip_atomic_fetch_add(p, v, __ATOMIC_RELAXED, __HIP_MEMORY_SCOPE_AGENT);
}

// ---------------------------------------------------------------------------
// h = x for rows < N_NODES, 0 for padding rows. float4 grid-stride.
// ---------------------------------------------------------------------------
__global__ void init_h_kernel(const float* __restrict__ x, float* __restrict__ h) {
  const int total = N_PAD * (HID / 4);
  for (int i = blockIdx.x * blockDim.x + threadIdx.x; i < total;
       i += gridDim.x * blockDim.x) {
    int row = i / (HID / 4);
    float4 v = make_float4(0.f, 0.f, 0.f, 0.f);
    if (row < N_NODES) v = ((const float4*)x)[i];
    ((float4*)h)[i] = v;
  }
}

// agg = h  (implements the "+ (1+eps)*x" self term, eps = 0)
__global__ void copy_kernel(const float* __restrict__ src, float* __restrict__ dst) {
  const int total = N_PAD * (HID / 4);
  for (int i = blockIdx.x * blockDim.x + threadIdx.x; i < total;
       i += gridDim.x * blockDim.x) {
    ((float4*)dst)[i] = ((const float4*)src)[i];
  }
}

__global__ void zero_kernel(float* __restrict__ p, int n) {
  for (int i = blockIdx.x * blockDim.x + threadIdx.x; i < n;
       i += gridDim.x * blockDim.x) p[i] = 0.f;
}

// ---------------------------------------------------------------------------
// Edge message + scatter: one wave32 per edge, 4 channels per lane.
// msg = relu(h[src] + relu(ew * edge_w + edge_b)); agg[dst] += msg (atomic f32)
// h[src] row load: 32 lanes x 16B = 512B fully coalesced.
// ---------------------------------------------------------------------------
__global__ __launch_bounds__(256)
void edge_msg_kernel(const float* __restrict__ h, float* __restrict__ agg,
                     const float* __restrict__ ew,        // [E]
                     const int* __restrict__ src_idx,     // [E]
                     const int* __restrict__ dst_idx,     // [E]
                     const float* __restrict__ edge_w,    // [HID] (this layer)
                     const float* __restrict__ edge_b) {  // [HID]
  int gtid = blockIdx.x * blockDim.x + threadIdx.x;
  int e    = gtid >> 5;
  int lane = gtid & 31;
  if (e >= N_EDGES) return;

  int s = src_idx[e];
  int d = dst_idx[e];
  float w = ew[e];

  int c0 = lane * 4;
  float4 hv = *(const float4*)(h + (size_t)s * HID + c0);
  float4 wv = *(const float4*)(edge_w + c0);
  float4 bv = *(const float4*)(edge_b + c0);

  float m0 = fmaxf(hv.x + fmaxf(w * wv.x + bv.x, 0.f), 0.f);
  float m1 = fmaxf(hv.y + fmaxf(w * wv.y + bv.y, 0.f), 0.f);
  float m2 = fmaxf(hv.z + fmaxf(w * wv.z + bv.z, 0.f), 0.f);
  float m3 = fmaxf(hv.w + fmaxf(w * wv.w + bv.w, 0.f), 0.f);

  float* out = agg + (size_t)d * HID + c0;
  atomic_add_f32(out + 0, m0);
  atomic_add_f32(out + 1, m1);
  atomic_add_f32(out + 2, m2);
  atomic_add_f32(out + 3, m3);
}

// ---------------------------------------------------------------------------
// Out = relu(A @ W^T) with V_WMMA_F32_16X16X4_F32.
// A: [N_PAD, HID] (agg + h, already combined). W: [HID, HID] row j = output j.
// Block = 256 threads = 8 waves; block covers 128 rows, wave covers 16 rows
// and all 8 column tiles (16 cols each). Weight staged in padded LDS.
// ---------------------------------------------------------------------------
__global__ __launch_bounds__(256)
void gemm_relu_wmma_kernel(const float* __restrict__ A,
                           const float* __restrict__ W,
                           float* __restrict__ Out) {
  __shared__ float lds[HID * LDSW];

  int tid = threadIdx.x;
  // Stage W into LDS (row-major, padded rows of LDSW floats).
  for (int i = tid * 4; i < HID * HID; i += 256 * 4) {
    float4 v = *(const float4*)(W + i);
    int row = i >> 7;
    int col = i & (HID - 1);
    float* p = &lds[row * LDSW + col];
    p[0] = v.x; p[1] = v.y; p[2] = v.z; p[3] = v.w;
  }
  __syncthreads();

  int waveId = tid >> 5;
  int lane   = tid & 31;
  int row0   = blockIdx.x * 128 + waveId * 16;

  // A fragment layout (32-bit A 16x4): lanes 0-15 hold M=lane, K={k,k+1};
  // lanes 16-31 hold M=lane-16, K={k+2,k+3}.
  int m    = lane & 15;
  int koff = (lane >> 4) << 1;          // 0 or 2

  const float* arow = A + (size_t)(row0 + m) * HID + koff;

  v8f acc[8] = {};

  for (int k = 0; k < HID; k += 4) {
    float2 av = *(const float2*)(arow + k);
    v2f a = {av.x, av.y};
#pragma unroll
    for (int t = 0; t < 8; ++t) {
      // B = W^T: B[kk][n] = W[n][kk]; mirror of the A striping.
      const float2 bvm = *(const float2*)(&lds[(t * 16 + m) * LDSW + k + koff]);
      v2f b = {bvm.x, bvm.y};
      acc[t] = __builtin_amdgcn_wmma_f32_16x16x4_f32(
          /*neg_a=*/false, a, /*neg_b=*/false, b,
          /*c_mod=*/(short)0, acc[t], /*reuse_a=*/false, /*reuse_b=*/false);
    }
  }

  // C/D layout: lanes 0-15 -> N=lane, M=v; lanes 16-31 -> N=lane-16, M=v+8.
  int mbase = (lane >> 4) << 3;         // 0 or 8
#pragma unroll
  for (int t = 0; t < 8; ++t) {
#pragma unroll
    for (int v = 0; v < 8; ++v) {
      float val = fmaxf(acc[t][v], 0.f);
      Out[(size_t)(row0 + mbase + v) * HID + t * 16 + m] = val;
    }
  }
}

// ---------------------------------------------------------------------------
// Mean-pool accumulate: one wave per node, 4 channels per lane.
// ---------------------------------------------------------------------------
__global__ __launch_bounds__(256)
void pool_accum_kernel(const float* __restrict__ h, const int* __restrict__ batch,
                       float* __restrict__ sums, float* __restrict__ cnt) {
  int gtid = blockIdx.x * blockDim.x + threadIdx.x;
  int node = gtid >> 5;
  int lane = gtid & 31;
  if (node >= N_NODES) return;
  int g = batch[node];
  int c0 = lane * 4;
  float4 hv = *(const float4*)(h + (size_t)node * HID + c0);
  float* out = sums + (size_t)g * HID + c0;
  atomic_add_f32(out + 0, hv.x);
  atomic_add_f32(out + 1, hv.y);
  atomic_add_f32(out + 2, hv.z);
  atomic_add_f32(out + 3, hv.w);
  if (lane == 0) atomic_add_f32(cnt + g, 1.0f);
}

// out[g][cls] = (sums[g] . readout_w[cls]) / max(cnt[g], 1)
__global__ void readout_kernel(const float* __restrict__ sums,
                               const float* __restrict__ cnt,
                               const float* __restrict__ rw,
                               float* __restrict__ out) {
  int g = blockIdx.x;
  int cls = threadIdx.x;
  if (g >= N_GRAPHS || cls >= N_CLASSES) return;
  float c = fmaxf(cnt[g], 1.0f);
  const float* s = sums + (size_t)g * HID;
  const float* w = rw + (size_t)cls * HID;
  float acc = 0.f;
  for (int k = 0; k < HID; ++k) acc += s[k] * w[k];
  out[g * N_CLASSES + cls] = acc / c;
}

// ---------------------------------------------------------------------------
extern "C" void kernel_launch(void* const* d_in, const int* in_sizes, int n_in,
                              void* d_out, int out_size, void* d_ws, size_t ws_size,
                              hipStream_t stream) {
  const float* x         = (const float*)d_in[0];   // [N_NODES, HID]
  const float* ew        = (const float*)d_in[1];   // [N_EDGES, 1]
  const float* lin_w     = (const float*)d_in[2];   // [L, HID, HID]
  const float* edge_w    = (const float*)d_in[3];   // [L, HID]
  const float* edge_b    = (const float*)d_in[4];   // [L, HID]
  const float* readout_w = (const float*)d_in[5];   // [N_CLASSES, HID]
  const int*   edge_idx  = (const int*)d_in[6];     // [2, N_EDGES]
  const int*   batch     = (const int*)d_in[7];     // [N_NODES]
  float* out = (float*)d_out;                       // [N_GRAPHS, N_CLASSES]

  const int* src_idx = edge_idx;
  const int* dst_idx = edge_idx + N_EDGES;

  // Workspace layout (floats): h[N_PAD*HID] | agg[N_PAD*HID] | sums[G*HID] | cnt[G]
  float* h_buf   = (float*)d_ws;
  float* agg_buf = h_buf + (size_t)N_PAD * HID;
  float* sums    = agg_buf + (size_t)N_PAD * HID;
  float* cnt     = sums + (size_t)N_GRAPHS * HID;

  const int copy_blocks = 2048;
  init_h_kernel<<<copy_blocks, 256, 0, stream>>>(x, h_buf);
  zero_kernel<<<64, 256, 0, stream>>>(sums, N_GRAPHS * HID + N_GRAPHS);

  const int edge_blocks = (N_EDGES + 7) / 8;   // 8 waves (edges) per 256-thr block
  const int gemm_blocks = N_PAD / 128;         // 391

  for (int l = 0; l < N_LAYERS; ++l) {
    copy_kernel<<<copy_blocks, 256, 0, stream>>>(h_buf, agg_buf);
    edge_msg_kernel<<<edge_blocks, 256, 0, stream>>>(
        h_buf, agg_buf, ew, src_idx, dst_idx,
        edge_w + (size_t)l * HID, edge_b + (size_t)l * HID);
    gemm_relu_wmma_kernel<<<gemm_blocks, 256, 0, stream>>>(
        agg_buf, lin_w + (size_t)l * HID * HID, h_buf);
  }

  const int pool_blocks = (N_NODES + 7) / 8;
  pool_accum_kernel<<<pool_blocks, 256, 0, stream>>>(h_buf, batch, sums, cnt);
  readout_kernel<<<N_GRAPHS, 32, 0, stream>>>(sums, cnt, readout_w, out);
}